// KeypointStreamEncoder_74938589381254
// MI455X (gfx1250) — compile-verified
//
#include <hip/hip_runtime.h>
#include <math.h>

// ---------------------------------------------------------------------------
// KeypointStreamEncoder on gfx1250 (MI455X).
// fp32 WMMA (v_wmma_f32_16x16x4_f32) for all GEMMs, big-LDS fused attention.
// ---------------------------------------------------------------------------

typedef float v2f __attribute__((ext_vector_type(2)));
typedef float v8f __attribute__((ext_vector_type(8)));

#define EDIM 512
#define JN 64
#define TNN 256
#define HEADS 8
#define HD 64
#define BTN 2048                 // B*T
#define ROWS 131072              // B*T*J
#define FMAXF 3.402823466e+38f
#define TINYF 1.17549435e-38f

#define XS_STRIDE 516            // 64x512 tile padded -> conflict-free b64 A loads
#define S_STRIDE 68              // 64x64 tiles padded

// LDS budgets (floats)
#define ATTN_LDS_FLOATS (64*XS_STRIDE + HEADS*64*S_STRIDE + 64*S_STRIDE + 64*S_STRIDE + 64 + 256 + 256) // 77120 -> 308480 B
#define PW_LDS_FLOATS   (64*XS_STRIDE + 256 + 256)                                                      // 33536 -> 134144 B

__device__ __forceinline__ v8f v8zero() {
  v8f r = {0.f,0.f,0.f,0.f,0.f,0.f,0.f,0.f};
  return r;
}

__device__ __forceinline__ v8f wmma4(v2f a, v2f b, v8f c) {
  // D = A(16x4) * B(4x16) + C(16x16), fp32
  return __builtin_amdgcn_wmma_f32_16x16x4_f32(false, a, false, b, (short)0, c, false, false);
}

// A fragment: row-major LDS tile. lane<16: rows m0+lane, K=k,k+1 ; lane>=16: K=k+2,k+3
__device__ __forceinline__ v2f ldA_lds(const float* base, int stride, int m0, int k, int lane) {
  int row = m0 + (lane & 15);
  int kb  = k + ((lane >> 4) << 1);
  const float* p = base + row * stride + kb;
  v2f r; r.x = p[0]; r.y = p[1];
  return r;
}

// B fragment from global row-major W[K][N]
__device__ __forceinline__ v2f ldB_g(const float* __restrict__ W, int ldw, int n0, int k, int lane) {
  int col = n0 + (lane & 15);
  int kb  = k + ((lane >> 4) << 1);
  v2f r;
  r.x = W[(size_t)kb * ldw + col];
  r.y = W[(size_t)(kb + 1) * ldw + col];
  return r;
}

// B fragment = (src)^T where src is row-major [N][K] in LDS  (B[d][n] = src[n][d])
__device__ __forceinline__ v2f ldB_t(const float* base, int stride, int n0, int k, int lane) {
  int col = n0 + (lane & 15);
  int kb  = k + ((lane >> 4) << 1);
  const float* p = base + col * stride + kb;
  v2f r; r.x = p[0]; r.y = p[1];
  return r;
}

// B fragment from row-major LDS tile [K][N]
__device__ __forceinline__ v2f ldB_lds(const float* base, int stride, int n0, int k, int lane) {
  int col = n0 + (lane & 15);
  int kb  = k + ((lane >> 4) << 1);
  v2f r;
  r.x = base[kb * stride + col];
  r.y = base[(kb + 1) * stride + col];
  return r;
}

// Store 16x16 C tile (v8f per lane) to LDS. VGPR r -> row m0 + r + (lane>=16)*8
__device__ __forceinline__ void stC_lds(float* base, int stride, int m0, int n0, int lane, v8f c) {
  int col = n0 + (lane & 15);
  int rb  = m0 + ((lane >> 4) << 3);
#pragma unroll
  for (int r = 0; r < 8; ++r) base[(rb + r) * stride + col] = c[r];
}

__device__ __forceinline__ float gelu_exact(float x) {
  return 0.5f * x * (1.f + erff(x * 0.70710678118654752f));
}

// ---------------------------------------------------------------------------
// Kernel 1: input LN(Din=6) -> @W_in -> + posenc(J,E)
// one block per row (b,t,j); 256 threads -> 2 output channels each
// ---------------------------------------------------------------------------
__global__ __launch_bounds__(256) void embed_kernel(
    const float* __restrict__ kp, const float* __restrict__ g6,
    const float* __restrict__ b6, const float* __restrict__ Win,
    float* __restrict__ X0) {
  __shared__ float kv[6];
  int row = blockIdx.x;
  int tid = threadIdx.x;
  if (tid < 6) kv[tid] = kp[(size_t)row * 6 + tid];
  __syncthreads();
  float m = 0.f;
#pragma unroll
  for (int d = 0; d < 6; ++d) m += kv[d];
  m *= (1.f / 6.f);
  float var = 0.f;
#pragma unroll
  for (int d = 0; d < 6; ++d) { float t = kv[d] - m; var += t * t; }
  var *= (1.f / 6.f);
  float rs = rsqrtf(var + 1e-5f);
  float xn[6];
#pragma unroll
  for (int d = 0; d < 6; ++d) xn[d] = (kv[d] - m) * rs * g6[d] + b6[d];
  int j = row & 63;
  const float kLog = -logf(10000.f) / (float)EDIM;
#pragma unroll
  for (int u = 0; u < 2; ++u) {
    int e = tid + u * 256;
    float div = expf(kLog * (float)(e & ~1));
    float arg = (float)j * div;
    float acc = (e & 1) ? cosf(arg) : sinf(arg);
#pragma unroll
    for (int d = 0; d < 6; ++d) acc += xn[d] * Win[d * EDIM + e];
    X0[(size_t)row * EDIM + e] = acc;
  }
}

// ---------------------------------------------------------------------------
// Kernel 2: fused tanh-attention + residual + LN.  One block per (b,t) frame.
// 256 threads (8 waves).  ~301 KB dynamic LDS.
// ---------------------------------------------------------------------------
__global__ __launch_bounds__(256) void attn_kernel(
    const float* __restrict__ X0, const unsigned char* __restrict__ msk,
    const float* __restrict__ Wq, const float* __restrict__ Wk,
    const float* __restrict__ Wv, const float* __restrict__ Wo,
    const float* __restrict__ gE, const float* __restrict__ bE,
    float* __restrict__ X1) {
  extern __shared__ float sm[];
  float* xs    = sm;                              // 64 x 516 input tile (also residual/out)
  float* sbuf  = xs + 64 * XS_STRIDE;             // 8 x (64 x 68) scores, later att
  float* tA    = sbuf + HEADS * 64 * S_STRIDE;    // 64 x 68 (q_h)
  float* tB    = tA + 64 * S_STRIDE;              // 64 x 68 (k_h / v_h)
  float* maskf = tB + 64 * S_STRIDE;              // 64
  float* red   = maskf + 64;                      // 256
  float* red2  = red + 256;                       // 256

  int bt   = blockIdx.x;
  int tid  = threadIdx.x;
  int lane = tid & 31;
  int wv   = tid >> 5;

  const float* Xrow = X0 + (size_t)bt * JN * EDIM;
  for (int i = tid; i < JN * (EDIM / 4); i += 256) {
    int r  = i >> 7;
    int c4 = (i & 127) << 2;
    float4 val = *(const float4*)(Xrow + (size_t)r * EDIM + c4);
    *(float4*)(xs + r * XS_STRIDE + c4) = val;
  }
  if (tid < JN) maskf[tid] = msk[(size_t)bt * JN + tid] ? 1.f : 0.f;
  __syncthreads();

  // 64x64 tile assignment: wave -> (mt, two ntiles)
  int mt   = (wv & 3) * 16;
  int ngrp = (wv >> 2) * 2;

  // ---------- pass 1: per head q,k then s = tanh(q k^T / 8) ----------
  for (int h = 0; h < HEADS; ++h) {
    const float* WqH = Wq + h * HD;
    const float* WkH = Wk + h * HD;
    v8f aq[2], ak[2];
    aq[0] = v8zero(); aq[1] = v8zero(); ak[0] = v8zero(); ak[1] = v8zero();
    for (int k = 0; k < EDIM; k += 4) {
      v2f a = ldA_lds(xs, XS_STRIDE, mt, k, lane);
#pragma unroll
      for (int t = 0; t < 2; ++t) {
        int n0 = (ngrp + t) * 16;
        v2f bq = ldB_g(WqH, EDIM, n0, k, lane);
        aq[t] = wmma4(a, bq, aq[t]);
        v2f bk = ldB_g(WkH, EDIM, n0, k, lane);
        ak[t] = wmma4(a, bk, ak[t]);
      }
    }
#pragma unroll
    for (int t = 0; t < 2; ++t) {
      stC_lds(tA, S_STRIDE, mt, (ngrp + t) * 16, lane, aq[t]);
      stC_lds(tB, S_STRIDE, mt, (ngrp + t) * 16, lane, ak[t]);
    }
    __syncthreads();
    v8f as[2];
    as[0] = v8zero(); as[1] = v8zero();
    for (int k = 0; k < HD; k += 4) {
      v2f a = ldA_lds(tA, S_STRIDE, mt, k, lane);
#pragma unroll
      for (int t = 0; t < 2; ++t) {
        v2f b = ldB_t(tB, S_STRIDE, (ngrp + t) * 16, k, lane);   // B = k^T
        as[t] = wmma4(a, b, as[t]);
      }
    }
    float* sh = sbuf + h * 64 * S_STRIDE;
#pragma unroll
    for (int t = 0; t < 2; ++t) {
      int col = (ngrp + t) * 16 + (lane & 15);
      int rb  = mt + ((lane >> 4) << 3);
#pragma unroll
      for (int r = 0; r < 8; ++r)
        sh[(rb + r) * S_STRIDE + col] = tanhf(as[t][r] * 0.125f);
    }
    __syncthreads();
  }

  // ---------- global softmax statistics over (H, q, k) ----------
  float lmax = -FMAXF;
  for (int idx = tid; idx < HEADS * JN * JN; idx += 256) {
    int h = idx >> 12;
    int q = (idx >> 6) & 63;
    int c = idx & 63;
    float mm = maskf[q] * maskf[c];
    float v  = sbuf[h * 64 * S_STRIDE + q * S_STRIDE + c];
    lmax = fmaxf(lmax, mm > 0.f ? v : -FMAXF);
  }
  red[tid] = lmax;
  __syncthreads();
  for (int s = 128; s > 0; s >>= 1) {
    if (tid < s) red[tid] = fmaxf(red[tid], red[tid + s]);
    __syncthreads();
  }
  float smax = red[0];
  __syncthreads();
  float lsum = 0.f;
  for (int idx = tid; idx < HEADS * JN * JN; idx += 256) {
    int h = idx >> 12;
    int q = (idx >> 6) & 63;
    int c = idx & 63;
    float mm = maskf[q] * maskf[c];
    float v  = sbuf[h * 64 * S_STRIDE + q * S_STRIDE + c];
    lsum += (mm > 0.f) ? expf(v - smax) : 0.f;
  }
  red[tid] = lsum;
  __syncthreads();
  for (int s = 128; s > 0; s >>= 1) {
    if (tid < s) red[tid] += red[tid + s];
    __syncthreads();
  }
  float invden = 1.f / fmaxf(red[0], TINYF);
  __syncthreads();

  // ---------- pass 2: per head v, att = w @ v (att overwrites score slab) ----------
  for (int h = 0; h < HEADS; ++h) {
    const float* WvH = Wv + h * HD;
    v8f av[2];
    av[0] = v8zero(); av[1] = v8zero();
    for (int k = 0; k < EDIM; k += 4) {
      v2f a = ldA_lds(xs, XS_STRIDE, mt, k, lane);
#pragma unroll
      for (int t = 0; t < 2; ++t) {
        v2f b = ldB_g(WvH, EDIM, (ngrp + t) * 16, k, lane);
        av[t] = wmma4(a, b, av[t]);
      }
    }
#pragma unroll
    for (int t = 0; t < 2; ++t) stC_lds(tB, S_STRIDE, mt, (ngrp + t) * 16, lane, av[t]);
    __syncthreads();

    float* sh = sbuf + h * 64 * S_STRIDE;
    v8f aa[2];
    aa[0] = v8zero(); aa[1] = v8zero();
    for (int k = 0; k < HD; k += 4) {
      // A fragment = softmax weights computed on the fly from stored scores
      int row = mt + (lane & 15);
      int kb  = k + ((lane >> 4) << 1);
      const float* p = sh + row * S_STRIDE + kb;
      float mr = maskf[row];
      float m0 = mr * maskf[kb];
      float m1 = mr * maskf[kb + 1];
      v2f a;
      a.x = (m0 > 0.f) ? expf(p[0] - smax) * invden : 0.f;
      a.y = (m1 > 0.f) ? expf(p[1] - smax) * invden : 0.f;
#pragma unroll
      for (int t = 0; t < 2; ++t) {
        v2f b = ldB_lds(tB, S_STRIDE, (ngrp + t) * 16, k, lane);
        aa[t] = wmma4(a, b, aa[t]);
      }
    }
    __syncthreads();   // all reads of sh done before overwrite
#pragma unroll
    for (int t = 0; t < 2; ++t) {
      int col = (ngrp + t) * 16 + (lane & 15);
      int rb  = mt + ((lane >> 4) << 3);
#pragma unroll
      for (int r = 0; r < 8; ++r)
        sh[(rb + r) * S_STRIDE + col] = aa[t][r] * maskf[rb + r];   // att *= mask_q
    }
    __syncthreads();
  }

  // ---------- out = att(64x512) @ Wo(512x512), residual into xs ----------
  int mtsBase = (wv >> 2) * 32;   // two m-tiles: mtsBase, mtsBase+16
  int ntBase  = (wv & 3) * 8;     // 8 n-tiles
  v8f acc[2][8];
#pragma unroll
  for (int t = 0; t < 2; ++t)
#pragma unroll
    for (int nn = 0; nn < 8; ++nn) acc[t][nn] = v8zero();
  for (int k = 0; k < EDIM; k += 4) {
    const float* sh = sbuf + (k >> 6) * 64 * S_STRIDE;
    int kk = k & 63;
    v2f a0 = ldA_lds(sh, S_STRIDE, mtsBase, kk, lane);
    v2f a1 = ldA_lds(sh, S_STRIDE, mtsBase + 16, kk, lane);
#pragma unroll
    for (int nn = 0; nn < 8; ++nn) {
      int n0 = (ntBase + nn) * 16;
      v2f b = ldB_g(Wo, EDIM, n0, k, lane);
      acc[0][nn] = wmma4(a0, b, acc[0][nn]);
      acc[1][nn] = wmma4(a1, b, acc[1][nn]);
    }
  }
#pragma unroll
  for (int t = 0; t < 2; ++t) {
    int mbase = mtsBase + t * 16 + ((lane >> 4) << 3);
#pragma unroll
    for (int nn = 0; nn < 8; ++nn) {
      int col = (ntBase + nn) * 16 + (lane & 15);
#pragma unroll
      for (int r = 0; r < 8; ++r)
        xs[(mbase + r) * XS_STRIDE + col] += acc[t][nn][r];
    }
  }
  __syncthreads();

  // ---------- row LayerNorm over E=512 + write ----------
  int rrow = tid >> 2;
  int part = tid & 3;
  float s1 = 0.f, s2 = 0.f;
  for (int c = part * 128; c < part * 128 + 128; ++c) {
    float v = xs[rrow * XS_STRIDE + c];
    s1 += v; s2 += v * v;
  }
  red[tid] = s1; red2[tid] = s2;
  __syncthreads();
  if (part == 0) {
    red[tid]  = red[tid] + red[tid + 1] + red[tid + 2] + red[tid + 3];
    red2[tid] = red2[tid] + red2[tid + 1] + red2[tid + 2] + red2[tid + 3];
  }
  __syncthreads();
  float mean = red[rrow << 2] * (1.f / 512.f);
  float var  = red2[rrow << 2] * (1.f / 512.f) - mean * mean;
  float rs   = rsqrtf(var + 1e-5f);
  float* outRow = X1 + ((size_t)bt * JN + rrow) * EDIM;
  for (int c = part * 128; c < part * 128 + 128; ++c) {
    float v = xs[rrow * XS_STRIDE + c];
    outRow[c] = (v - mean) * rs * gE[c] + bE[c];
  }
}

// ---------------------------------------------------------------------------
// Kernel 3: depthwise temporal conv (k=3, pad=1) + exact GELU (elementwise)
// ---------------------------------------------------------------------------
__global__ __launch_bounds__(256) void dwconv_kernel(
    const float* __restrict__ x, const float* __restrict__ dw, float* __restrict__ y) {
  int idx = blockIdx.x * 256 + threadIdx.x;          // 16,777,216 float4 items
  int e4 = (idx & 127) << 2;
  int j  = (idx >> 7) & 63;
  int t  = (idx >> 13) & 255;
  int b  = idx >> 21;
  long rowBase = (((long)b * TNN + t) * JN + j) * EDIM + e4;
  const long strideT = (long)JN * EDIM;
  float a0 = 0.f, a1 = 0.f, a2 = 0.f, a3 = 0.f;
#pragma unroll
  for (int kk = 0; kk < 3; ++kk) {
    int tt = t + kk - 1;
    if (tt < 0 || tt >= TNN) continue;
    float4 xv = *(const float4*)(x + rowBase + (long)(kk - 1) * strideT);
    float4 wv = *(const float4*)(dw + kk * EDIM + e4);
    a0 += xv.x * wv.x; a1 += xv.y * wv.y; a2 += xv.z * wv.z; a3 += xv.w * wv.w;
  }
  float4 o;
  o.x = gelu_exact(a0); o.y = gelu_exact(a1); o.z = gelu_exact(a2); o.w = gelu_exact(a3);
  *(float4*)(y + rowBase) = o;
}

// ---------------------------------------------------------------------------
// Kernel 4: pointwise 512x512 GEMM + residual + LayerNorm.
// One block per 64-row tile.  In-place safe vs Cin (tile read fully first).
// ---------------------------------------------------------------------------
__global__ __launch_bounds__(256) void pwln_kernel(
    const float* __restrict__ Cin, const float* __restrict__ Xres,
    const float* __restrict__ W, const float* __restrict__ gE,
    const float* __restrict__ bE, float* __restrict__ Out) {
  extern __shared__ float sm[];
  float* ct   = sm;                   // 64 x 516
  float* red  = ct + 64 * XS_STRIDE;  // 256
  float* red2 = red + 256;            // 256

  size_t rowOff = (size_t)blockIdx.x * 64;
  int tid  = threadIdx.x;
  int lane = tid & 31;
  int wv   = tid >> 5;

  for (int i = tid; i < 64 * 128; i += 256) {
    int r  = i >> 7;
    int c4 = (i & 127) << 2;
    *(float4*)(ct + r * XS_STRIDE + c4) = *(const float4*)(Cin + (rowOff + r) * EDIM + c4);
  }
  __syncthreads();

  int mtsBase = (wv >> 2) * 32;
  int ntBase  = (wv & 3) * 8;
  v8f acc[2][8];
#pragma unroll
  for (int t = 0; t < 2; ++t)
#pragma unroll
    for (int nn = 0; nn < 8; ++nn) acc[t][nn] = v8zero();
  for (int k = 0; k < EDIM; k += 4) {
    v2f a0 = ldA_lds(ct, XS_STRIDE, mtsBase, k, lane);
    v2f a1 = ldA_lds(ct, XS_STRIDE, mtsBase + 16, k, lane);
#pragma unroll
    for (int nn = 0; nn < 8; ++nn) {
      int n0 = (ntBase + nn) * 16;
      v2f b = ldB_g(W, EDIM, n0, k, lane);
      acc[0][nn] = wmma4(a0, b, acc[0][nn]);
      acc[1][nn] = wmma4(a1, b, acc[1][nn]);
    }
  }
  __syncthreads();   // all A reads done before overwriting ct

#pragma unroll
  for (int t = 0; t < 2; ++t) {
    int mbase = mtsBase + t * 16 + ((lane >> 4) << 3);
#pragma unroll
    for (int nn = 0; nn < 8; ++nn) {
      int col = (ntBase + nn) * 16 + (lane & 15);
#pragma unroll
      for (int r = 0; r < 8; ++r) {
        int row = mbase + r;
        float v = acc[t][nn][r] + Xres[(rowOff + row) * EDIM + col];
        ct[row * XS_STRIDE + col] = v;
      }
    }
  }
  __syncthreads();

  int rrow = tid >> 2;
  int part = tid & 3;
  float s1 = 0.f, s2 = 0.f;
  for (int c = part * 128; c < part * 128 + 128; ++c) {
    float v = ct[rrow * XS_STRIDE + c];
    s1 += v; s2 += v * v;
  }
  red[tid] = s1; red2[tid] = s2;
  __syncthreads();
  if (part == 0) {
    red[tid]  = red[tid] + red[tid + 1] + red[tid + 2] + red[tid + 3];
    red2[tid] = red2[tid] + red2[tid + 1] + red2[tid + 2] + red2[tid + 3];
  }
  __syncthreads();
  float mean = red[rrow << 2] * (1.f / 512.f);
  float var  = red2[rrow << 2] * (1.f / 512.f) - mean * mean;
  float rs   = rsqrtf(var + 1e-5f);
  float* outRow = Out + (rowOff + rrow) * EDIM;
  for (int c = part * 128; c < part * 128 + 128; ++c) {
    float v = ct[rrow * XS_STRIDE + c];
    outRow[c] = (v - mean) * rs * gE[c] + bE[c];
  }
}

// ---------------------------------------------------------------------------
// Kernel 5: frame = masked mean over joints
// ---------------------------------------------------------------------------
__global__ __launch_bounds__(256) void frame_kernel(
    const float* __restrict__ joint, const unsigned char* __restrict__ msk,
    float* __restrict__ frame) {
  int bt  = blockIdx.x;
  int tid = threadIdx.x;
  int c0 = tid, c1 = tid + 256;
  const float* base = joint + (size_t)bt * JN * EDIM;
  float msum = 0.f, acc0 = 0.f, acc1 = 0.f;
  for (int j = 0; j < JN; ++j) {
    float mf = msk[(size_t)bt * JN + j] ? 1.f : 0.f;
    msum += mf;
    acc0 += mf * base[(size_t)j * EDIM + c0];
    acc1 += mf * base[(size_t)j * EDIM + c1];
  }
  float denom = fmaxf(msum, 1.f);
  frame[(size_t)bt * EDIM + c0] = acc0 / denom;
  frame[(size_t)bt * EDIM + c1] = acc1 / denom;
}

// ---------------------------------------------------------------------------
extern "C" void kernel_launch(void* const* d_in, const int* in_sizes, int n_in,
                              void* d_out, int out_size, void* d_ws, size_t ws_size,
                              hipStream_t stream) {
  (void)in_sizes; (void)n_in; (void)out_size; (void)ws_size;
  const float* kp      = (const float*)d_in[0];
  const unsigned char* msk = (const unsigned char*)d_in[1];   // jax bool -> 1 byte
  const float* ln_in_g = (const float*)d_in[2];
  const float* ln_in_b = (const float*)d_in[3];
  const float* W_in    = (const float*)d_in[4];
  const float* Wq      = (const float*)d_in[5];
  const float* Wk      = (const float*)d_in[6];
  const float* Wv      = (const float*)d_in[7];
  const float* Wo      = (const float*)d_in[8];
  const float* an_g    = (const float*)d_in[9];
  const float* an_b    = (const float*)d_in[10];
  const float* dw0     = (const float*)d_in[11];
  const float* pw0     = (const float*)d_in[12];
  const float* n0_g    = (const float*)d_in[13];
  const float* n0_b    = (const float*)d_in[14];
  const float* dw1     = (const float*)d_in[15];
  const float* pw1     = (const float*)d_in[16];
  const float* n1_g    = (const float*)d_in[17];
  const float* n1_b    = (const float*)d_in[18];

  float* X0 = (float*)d_ws;                       // 268 MB
  float* X1 = X0 + (size_t)ROWS * EDIM;           // 268 MB
  float* joint = (float*)d_out;                   // also reused as conv scratch
  float* frame = joint + (size_t)ROWS * EDIM;

  size_t attnLds = (size_t)ATTN_LDS_FLOATS * sizeof(float);
  size_t pwLds   = (size_t)PW_LDS_FLOATS * sizeof(float);
  (void)hipFuncSetAttribute(reinterpret_cast<const void*>(attn_kernel),
                            hipFuncAttributeMaxDynamicSharedMemorySize, (int)attnLds);
  (void)hipFuncSetAttribute(reinterpret_cast<const void*>(pwln_kernel),
                            hipFuncAttributeMaxDynamicSharedMemorySize, (int)pwLds);

  // x0 = LN(kp) @ W_in + posenc
  embed_kernel<<<ROWS, 256, 0, stream>>>(kp, ln_in_g, ln_in_b, W_in, X0);
  // x1 = LN(x0 + attn(x0))
  attn_kernel<<<BTN, 256, attnLds, stream>>>(X0, msk, Wq, Wk, Wv, Wo, an_g, an_b, X1);
  // block 0: joint-region used as conv scratch
  dwconv_kernel<<<(ROWS * (EDIM / 4)) / 256, 256, 0, stream>>>(X1, dw0, joint);
  pwln_kernel<<<BTN, 256, pwLds, stream>>>(joint, X1, pw0, n0_g, n0_b, X0);
  // block 1: write final joint output (in-place safe per 64-row tile)
  dwconv_kernel<<<(ROWS * (EDIM / 4)) / 256, 256, 0, stream>>>(X0, dw1, joint);
  pwln_kernel<<<BTN, 256, pwLds, stream>>>(joint, X0, pw1, n1_g, n1_b, joint);
  // frame = masked mean over joints
  frame_kernel<<<BTN, 256, 0, stream>>>(joint, msk, frame);
}